// QuantumInterference_38826504356538
// MI455X (gfx1250) — compile-verified
//
#include <hip/hip_runtime.h>
#include <math.h>

typedef float v2f __attribute__((ext_vector_type(2)));
typedef float v4f __attribute__((ext_vector_type(4)));
typedef float v8f __attribute__((ext_vector_type(8)));

#define B_DIM 32
#define D_DIM 1024
#define RWAVES 8                    // waves per reduce block (K-slices)

// ---------------------------------------------------------------------------
// Phase 1: per-batch-row reductions Cs_b = sum_j re/r, Ss_b = sum_j im/r,
// Rsq_b = sum_j (re^2+im^2), as GEMV [16 x 1024] * ones(1024) on the matrix
// pipe (V_WMMA_F32_16X16X4_F32).  grid = 2 blocks (16 batch rows each),
// block = 256 threads = 8 waves; wave w reduces K-slice [w*128, w*128+128)
// (32 WMMA K-steps), partials combined through LDS.
// A-layout per ISA: lanes 0-15 feed K=k0,k0+1 (VGPR0/1), lanes 16-31 feed
// K=k0+2,k0+3.  EXEC is all ones for every wave throughout the WMMA loop.
// ---------------------------------------------------------------------------
__global__ void __launch_bounds__(256)
qi_reduce_kernel(const float* __restrict__ re,
                 const float* __restrict__ im,
                 float* __restrict__ sums /* [Cs x32][Ss x32][Rsq x32] */) {
  __shared__ float part[3][16][RWAVES];   // [quantity][row][wave]

  const int tid   = threadIdx.x;
  const int lane  = tid & 31;
  const int w     = tid >> 5;             // wave id 0..7
  const int tile  = blockIdx.x;           // 0..1 -> batch rows tile*16..+15
  const int m     = lane & 15;            // A-matrix row this lane feeds
  const int khalf = (lane >> 4) * 2;
  const int b     = tile * 16 + m;
  const float* rowr = re + b * D_DIM;
  const float* rowi = im + b * D_DIM;
  const int kbase = w * (D_DIM / RWAVES); // 128 K-columns per wave

  v2f bones; bones[0] = 1.0f; bones[1] = 1.0f;   // B = ones(4x16)
  v8f accC = {}; v8f accS = {}; v8f accR = {};

  for (int k0 = 0; k0 < D_DIM / RWAVES; k0 += 4) {
    const int k = kbase + k0 + khalf;
    v2f r2 = *(const v2f*)(rowr + k);     // elements k, k+1 (8B aligned)
    v2f i2 = *(const v2f*)(rowi + k);
    v2f mag, c, s;
    mag[0] = r2[0]*r2[0] + i2[0]*i2[0];
    mag[1] = r2[1]*r2[1] + i2[1]*i2[1];
    const float inv0 = __builtin_amdgcn_rsqf(mag[0]);  // v_rsq_f32 (~1 ulp)
    const float inv1 = __builtin_amdgcn_rsqf(mag[1]);
    c[0] = r2[0]*inv0; c[1] = r2[1]*inv1;              // cos(atan2) = re/r
    s[0] = i2[0]*inv0; s[1] = i2[1]*inv1;              // sin(atan2) = im/r
    // (neg_a, A, neg_b, B, c_mod, C, reuse_a, reuse_b)
    accC = __builtin_amdgcn_wmma_f32_16x16x4_f32(false, c,   false, bones, (short)0, accC, false, false);
    accS = __builtin_amdgcn_wmma_f32_16x16x4_f32(false, s,   false, bones, (short)0, accS, false, false);
    accR = __builtin_amdgcn_wmma_f32_16x16x4_f32(false, mag, false, bones, (short)0, accR, false, false);
  }

  // C/D layout: lanes 0-15 hold M = vgpr index; lanes 16-31 hold M = 8+vgpr.
  // All 16 columns equal the row sum (B was all-ones) -> read col 0 / 16.
  if (lane == 0) {
#pragma unroll
    for (int v = 0; v < 8; ++v) {
      part[0][v][w] = accC[v];
      part[1][v][w] = accS[v];
      part[2][v][w] = accR[v];
    }
  } else if (lane == 16) {
#pragma unroll
    for (int v = 0; v < 8; ++v) {
      part[0][8 + v][w] = accC[v];
      part[1][8 + v][w] = accS[v];
      part[2][8 + v][w] = accR[v];
    }
  }
  __syncthreads();

  if (tid < 48) {                          // 3 quantities x 16 rows
    const int q  = tid >> 4;               // 0..2
    const int mm = tid & 15;               // row within tile
    float acc = 0.0f;
#pragma unroll
    for (int ww = 0; ww < RWAVES; ++ww) acc += part[q][mm][ww];
    sums[q * 32 + tile * 16 + mm] = acc;
  }
}

// ---------------------------------------------------------------------------
// Phase 2 (the whole cost): out[b,i,j,{0,1}] = {re[b,j], im[b,j]} * t[b,i]
// with t = sign(S[b,i]) / R_b, computed inline from uniform scalar loads
// (sign(S) == sign(re*Cs + im*Ss) since r_i > 0).
// 256 MiB streaming output -> non-temporal b128 stores; row inputs are
// L2/L1 resident (8 KB per b, reused 1024x).  One block per (b,i) row.
// ---------------------------------------------------------------------------
__global__ void __launch_bounds__(256)
qi_write_kernel(const float* __restrict__ re,
                const float* __restrict__ im,
                const float* __restrict__ sums,
                float* __restrict__ out) {
  const int row = blockIdx.x;                 // row = b*D + i  (uniform)
  const int b   = row >> 10;
  const float Cs   = sums[b];                 // uniform -> s_load
  const float Ss   = sums[32 + b];
  const float Rsq  = sums[64 + b];
  const float d    = re[row] * Cs + im[row] * Ss;
  const float invR = __builtin_amdgcn_rsqf(Rsq);
  const float t    = (d >= 0.0f) ? invR : -invR;

  const float* rowr = re + (b << 10);
  const float* rowi = im + (b << 10);
  v4f* orow = (v4f*)(out + (size_t)row * (size_t)(2 * D_DIM));
#pragma unroll
  for (int u = 0; u < 2; ++u) {
    const int q = threadIdx.x + u * 256;      // float4 index 0..511 (j = 2q, 2q+1)
    v2f r2 = *(const v2f*)(rowr + 2 * q);
    v2f i2 = *(const v2f*)(rowi + 2 * q);
    v4f o;
    o[0] = r2[0] * t;   // real, j = 2q
    o[1] = i2[0] * t;   // imag, j = 2q
    o[2] = r2[1] * t;   // real, j = 2q+1
    o[3] = i2[1] * t;   // imag, j = 2q+1
    __builtin_nontemporal_store(o, orow + q);
  }
}

// ---------------------------------------------------------------------------
extern "C" void kernel_launch(void* const* d_in, const int* in_sizes, int n_in,
                              void* d_out, int out_size, void* d_ws, size_t ws_size,
                              hipStream_t stream) {
  (void)in_sizes; (void)n_in; (void)out_size; (void)ws_size;
  const float* re = (const float*)d_in[0];   // states_real [B,D] f32
  const float* im = (const float*)d_in[1];   // states_imag [B,D] f32
  float* out  = (float*)d_out;               // [B,D,D,2] f32
  float* sums = (float*)d_ws;                // 96 floats: Cs[32] Ss[32] Rsq[32]

  qi_reduce_kernel<<<dim3(2), dim3(256), 0, stream>>>(re, im, sums);
  qi_write_kernel<<<dim3(B_DIM * D_DIM), dim3(256), 0, stream>>>(re, im, sums, out);
}